// Reconstructeur2_66812511256935
// MI455X (gfx1250) — compile-verified
//
#include <hip/hip_runtime.h>
#include <hip/hip_bf16.h>

// Problem constants (from reference)
#define LATENT   1024
#define N_MLP    64
#define N_GRID2  16384          // 128*128 grid points
#define OUT_DIM  3
#define NJ       192            // N_MLP * OUT_DIM flattened output index j = m*3+o
#define W_ROWSTRIDE 3078        // (LATENT+2)*OUT_DIM floats per decoder
#define KCHUNKS  64             // K=1024 split into 64 chunks of 16
#define JT_TILES 12             // 192 / 16

typedef __attribute__((ext_vector_type(2))) float v2f;
typedef __attribute__((ext_vector_type(8))) float v8f;

// Hardware tanh (V_TANH_F32) when available on this toolchain; ocml fallback.
#if defined(__has_builtin)
#if __has_builtin(__builtin_amdgcn_tanhf)
#define FAST_TANH(v) __builtin_amdgcn_tanhf(v)
#endif
#endif
#ifndef FAST_TANH
#define FAST_TANH(v) tanhf(v)
#endif

// ---------------------------------------------------------------------------
// Phase 1: partial[kc*192 + j] = sum_{k in chunk kc} x[k] * W[m, k, o]
// One wave per (jt, kc). Uses V_WMMA_F32_16X16X4_F32 (full fp32 accumulate).
//
// Mapping: D = A x B with
//   A[p, q] = W^T[jt*16+p, kbase+q]   (16 output rows x 4 K)
//   B[q, n] = x[kbase+q]  (broadcast over all 16 columns n)
//   => D[p, n] = partial R[jt*16+p] in every column; read column 0.
//
// Per ISA layouts (32-bit A 16x4 and B 4x16):
//   lane L (L<16):  v0 -> K=0, v1 -> K=1 ; lane L (>=16): v0 -> K=2, v1 -> K=3
//   A row index = lane & 15.
//   D: VGPR v, lanes 0-15 = (M=v, N=lane); lanes 16-31 = (M=v+8, N=lane-16).
// ---------------------------------------------------------------------------
__global__ __launch_bounds__(32)
void phase1_wmma(const float* __restrict__ x,
                 const float* __restrict__ W,
                 float* __restrict__ partial) {
    const int jt   = (int)blockIdx.x >> 6;      // 0..11
    const int kc   = (int)blockIdx.x & 63;      // 0..63
    const int lane = (int)threadIdx.x;          // 0..31
    const int row  = lane & 15;
    const int j    = jt * 16 + row;             // flattened output index
    const int m    = j / 3;
    const int o    = j - 3 * m;
    const float* wrow = W + (size_t)m * W_ROWSTRIDE + o;  // index by 3*k
    const int koff = (lane >> 4) << 1;          // 0 for lanes 0-15, 2 for 16-31
    const int kbase = kc * 16;

    v8f acc = {};
    #pragma unroll
    for (int s = 0; s < 4; ++s) {
        const int k0 = kbase + s * 4 + koff;
        v2f a, b;
        a.x = wrow[3 * k0];
        a.y = wrow[3 * (k0 + 1)];
        b.x = x[k0];
        b.y = x[k0 + 1];
        acc = __builtin_amdgcn_wmma_f32_16x16x4_f32(
            /*neg_a=*/false, a, /*neg_b=*/false, b,
            /*c_mod=*/(short)0, acc, /*reuse_a=*/false, /*reuse_b=*/false);
    }

    // Column 0 of D lives in lane 0 (rows 0-7) and lane 16 (rows 8-15).
    float* p = partial + (size_t)kc * NJ + jt * 16;
    if (lane == 0) {
        #pragma unroll
        for (int v = 0; v < 8; ++v) p[v] = acc[v];
    } else if (lane == 16) {
        #pragma unroll
        for (int v = 0; v < 8; ++v) p[8 + v] = acc[v];
    }
}

// ---------------------------------------------------------------------------
// Reduce: coef[j] = b[j] + sum_kc partial[kc*192+j] ; also extract Wx, Wy.
// coef layout: [0..191]=c, [192..383]=wx, [384..575]=wy
// ---------------------------------------------------------------------------
__global__ __launch_bounds__(256)
void reduce_coef(const float* __restrict__ partial,
                 const float* __restrict__ b,
                 const float* __restrict__ W,
                 float* __restrict__ coef) {
    const int j = (int)threadIdx.x;
    if (j < NJ) {
        float s = b[j];
        #pragma unroll 8
        for (int kc = 0; kc < KCHUNKS; ++kc)
            s += partial[(size_t)kc * NJ + j];
        const int m = j / 3;
        const int o = j - 3 * m;
        coef[j]           = s;
        coef[NJ + j]      = W[(size_t)m * W_ROWSTRIDE + 3 * LATENT + o];       // W[m,1024,o]
        coef[2 * NJ + j]  = W[(size_t)m * W_ROWSTRIDE + 3 * (LATENT + 1) + o]; // W[m,1025,o]
    }
}

// ---------------------------------------------------------------------------
// Phase 2 (store-bandwidth-bound): out[(m*G+g)*3+o] = tanh(c + gx*wx + gy*wy)
// Each thread: 4 consecutive grid points of one decoder m.
//   loads : 2x float4 (grid, 32B aligned)    stores: 3x float4 (48B contiguous)
// ---------------------------------------------------------------------------
__global__ __launch_bounds__(256)
void phase2_eval(const float* __restrict__ coef,
                 const float* __restrict__ grid,
                 float* __restrict__ out) {
    const int tid = (int)blockIdx.x * 256 + (int)threadIdx.x;  // 262144 total
    const int m   = tid >> 12;          // 4096 quads per decoder
    const int q   = tid & 4095;
    const int g0  = q << 2;

    const float4* gp = (const float4*)(grid + (size_t)g0 * 2);
    const float4 gA = gp[0];   // X[g0],Y[g0],X[g0+1],Y[g0+1]
    const float4 gB = gp[1];   // X[g0+2],Y[g0+2],X[g0+3],Y[g0+3]

    const float c0  = coef[m * 3 + 0], c1  = coef[m * 3 + 1], c2  = coef[m * 3 + 2];
    const float wx0 = coef[NJ + m * 3 + 0], wx1 = coef[NJ + m * 3 + 1], wx2 = coef[NJ + m * 3 + 2];
    const float wy0 = coef[2 * NJ + m * 3 + 0], wy1 = coef[2 * NJ + m * 3 + 1], wy2 = coef[2 * NJ + m * 3 + 2];

    const float r0  = FAST_TANH(c0 + gA.x * wx0 + gA.y * wy0);
    const float r1  = FAST_TANH(c1 + gA.x * wx1 + gA.y * wy1);
    const float r2  = FAST_TANH(c2 + gA.x * wx2 + gA.y * wy2);
    const float r3  = FAST_TANH(c0 + gA.z * wx0 + gA.w * wy0);
    const float r4  = FAST_TANH(c1 + gA.z * wx1 + gA.w * wy1);
    const float r5  = FAST_TANH(c2 + gA.z * wx2 + gA.w * wy2);
    const float r6  = FAST_TANH(c0 + gB.x * wx0 + gB.y * wy0);
    const float r7  = FAST_TANH(c1 + gB.x * wx1 + gB.y * wy1);
    const float r8  = FAST_TANH(c2 + gB.x * wx2 + gB.y * wy2);
    const float r9  = FAST_TANH(c0 + gB.z * wx0 + gB.w * wy0);
    const float r10 = FAST_TANH(c1 + gB.z * wx1 + gB.w * wy1);
    const float r11 = FAST_TANH(c2 + gB.z * wx2 + gB.w * wy2);

    float4* op = (float4*)(out + ((size_t)m * N_GRID2 + g0) * 3);  // 48B aligned
    op[0] = make_float4(r0, r1, r2, r3);
    op[1] = make_float4(r4, r5, r6, r7);
    op[2] = make_float4(r8, r9, r10, r11);
}

extern "C" void kernel_launch(void* const* d_in, const int* in_sizes, int n_in,
                              void* d_out, int out_size, void* d_ws, size_t ws_size,
                              hipStream_t stream) {
    const float* x    = (const float*)d_in[0];   // [1024]
    const float* W    = (const float*)d_in[1];   // [64, 1026, 3]
    const float* b    = (const float*)d_in[2];   // [64, 3]
    const float* grid = (const float*)d_in[3];   // [16384, 2]
    float* out        = (float*)d_out;           // [64*16384, 3]

    float* coef    = (float*)d_ws;               // 576 floats
    float* partial = coef + 3 * NJ;              // 64*192 floats

    // Phase 1: 12 j-tiles x 64 K-chunks, one wave32 each.
    phase1_wmma<<<dim3(JT_TILES * KCHUNKS), dim3(32), 0, stream>>>(x, W, partial);
    // Deterministic reduction + Wx/Wy extraction.
    reduce_coef<<<dim3(1), dim3(256), 0, stream>>>(partial, b, W, coef);
    // Phase 2: 64*16384/4 threads.
    phase2_eval<<<dim3((N_MLP * N_GRID2 / 4) / 256), dim3(256), 0, stream>>>(coef, grid, out);
}